// Split_31714038514238
// MI455X (gfx1250) — compile-verified
//
#include <hip/hip_runtime.h>
#include <hip/hip_bf16.h>

// ---------------------------------------------------------------------------
// Problem:  indices[16384] in [0,64), z[1024] f32, W[64,1024,1024] f32,
//           b[64,1024] f32  ->  out[16384,1024] f32
// Phase 1:  expert_out[e,m] = dot(W[e,m,:], z) + b[e,m]   (WMMA f32 16x16x4)
// Phase 2:  out[i,:] = expert_out[indices[i],:]           (streaming gather)
// Roofline: 256MB W read (~11us) + 64MB out write (~3us) @ 23.3 TB/s.
// ---------------------------------------------------------------------------

typedef __attribute__((ext_vector_type(2))) float v2f;
typedef __attribute__((ext_vector_type(4))) float v4f;
typedef __attribute__((ext_vector_type(8))) float v8f;

#define Z_DIM      1024
#define N_EXPERTS  64
#define BATCH      16384
#define WAVES_PER_BLOCK 8   // 256 threads, wave32

// Each wave computes 16 consecutive rows of the flattened (65536 x 1024)
// expert-weight matrix times z, using V_WMMA_F32_16X16X4_F32 chained over K.
// B operand = z broadcast across all 16 N columns, so every column of the
// 16x16 accumulator carries the same dot product.
__global__ void __launch_bounds__(256)
expert_gemv_wmma(const float* __restrict__ W,
                 const float* __restrict__ z,
                 const float* __restrict__ bias,
                 float* __restrict__ eout) {
    __shared__ float zs[Z_DIM];

    // Stage z in LDS (4 KB) once per block.
    for (int i = threadIdx.x; i < Z_DIM; i += blockDim.x)
        zs[i] = z[i];
    __syncthreads();

    const int lane = threadIdx.x & 31;
    const int wave = threadIdx.x >> 5;

    // Global row block for this wave: 16 rows of the 65536-row matrix.
    const size_t row0  = ((size_t)blockIdx.x * WAVES_PER_BLOCK + wave) * 16;
    const int    mrow  = lane & 15;            // M within the 16-row tile
    const int    khalf = (lane >> 4) * 2;      // lanes 0-15: K+0,K+1 ; lanes 16-31: K+2,K+3

    // A-fragment source: this lane walks row (row0+mrow), starting at khalf.
    const float* __restrict__ wrow = W + (row0 + (size_t)mrow) * Z_DIM + khalf;
    const float* __restrict__ zrow = zs + khalf;

    v8f acc = {};                              // 16x16 f32 C/D fragment
    for (int k0 = 0; k0 < Z_DIM; k0 += 16) {
        // Keep the HBM stream ahead of the WMMAs (global_prefetch_b8).
        __builtin_prefetch(wrow + k0 + 512, 0, 0);
#pragma unroll
        for (int kk = 0; kk < 16; kk += 4) {
            // A: 16x4 tile of W (this lane: 2 consecutive f32 of its row)
            v2f a = *(const v2f*)(wrow + k0 + kk);
            // B: 4x16 tile = z[k0+kk .. k0+kk+3] broadcast over N (from LDS)
            v2f bz = *(const v2f*)(zrow + k0 + kk);
            acc = __builtin_amdgcn_wmma_f32_16x16x4_f32(
                /*neg_a=*/false, a, /*neg_b=*/false, bz,
                /*c_mod=*/(short)0, acc, /*reuse_a=*/false, /*reuse_b=*/false);
        }
    }

    // C/D layout: VGPR v -> M=v (lanes 0-15) / M=8+v (lanes 16-31); all N
    // columns identical (B was broadcast).  Lane 0 writes rows m0..m0+7,
    // lane 16 writes rows m0+8..m0+15.  16 floats per wave total.
    if (lane == 0 || lane == 16) {
        const size_t base = row0 + (size_t)(lane >> 4) * 8;
#pragma unroll
        for (int v = 0; v < 8; ++v)
            eout[base + v] = acc[v] + bias[base + v];
    }
}

// One block per output row: 256 threads x v4f = 4 KB row copy.
// expert_out (256 KB) stays L2-resident; the 64 MB output stream is written
// with nontemporal b128 stores so it does not evict it.
__global__ void __launch_bounds__(256)
gather_rows(const int* __restrict__ idx,
            const float* __restrict__ eout,
            float* __restrict__ out) {
    const int row = blockIdx.x;
    const int e   = idx[row];
    const v4f* __restrict__ src = (const v4f*)(eout + (size_t)e   * Z_DIM);
    v4f*       __restrict__ dst = (v4f*)      (out  + (size_t)row * Z_DIM);
    const v4f v = src[threadIdx.x];
    __builtin_nontemporal_store(v, dst + threadIdx.x);
}

extern "C" void kernel_launch(void* const* d_in, const int* in_sizes, int n_in,
                              void* d_out, int out_size, void* d_ws, size_t ws_size,
                              hipStream_t stream) {
    const int*   idx = (const int*)  d_in[0];   // indices [16384]
    const float* z   = (const float*)d_in[1];   // z       [1024]
    const float* W   = (const float*)d_in[2];   // W       [64,1024,1024]
    const float* b   = (const float*)d_in[3];   // b       [64,1024]
    float*       out = (float*)d_out;           // out     [16384,1024]
    float*       eout = (float*)d_ws;           // scratch: 64*1024 f32 = 256 KB

    // Phase 1: 65536 rows / (16 rows/wave * 8 waves/block) = 512 blocks.
    expert_gemv_wmma<<<(N_EXPERTS * Z_DIM) / (16 * WAVES_PER_BLOCK), 256, 0, stream>>>(
        W, z, b, eout);

    // Phase 2: one block per batch row.
    gather_rows<<<BATCH, 256, 0, stream>>>(idx, eout, out);
}